// BoxCrossCategoryLoss_68745246539808
// MI455X (gfx1250) — compile-verified
//
#include <hip/hip_runtime.h>
#include <hip/hip_bf16.h>
#include <stdint.h>

// ---------------------------------------------------------------------------
// BoxCrossCategoryLoss for MI455X (gfx1250, wave32)
//
// Streaming elementwise reduction: 48 B/row (6 x float2), 15 log1mexp/row,
// 36 relu terms, global scalar sum.
//   HBM floor: 384 MB / 23.3 TB/s ~= 16.5 us; transcendental cost comparable.
// CDNA5 path: global_load_async_to_lds_b128 double-buffered staging
// (ASYNCcnt / s_wait_asynccnt), ds_load_b128 consumption, wave32 shuffle
// reduction, deterministic two-stage final sum (no float atomics).
// ---------------------------------------------------------------------------

#define WG_BLOCKS  2048
#define WG_THREADS 256
#define TILE       WG_THREADS            // float4 units per tile per block

// Numerically stable log(1 - exp(x)) for x < 0 (branchless, safe operands).
__device__ __forceinline__ float log1mexp_f(float x) {
    const bool  big = x > -0.69314718f;          // x > log(1/2)
    const float xb  = big ? x : -1.0f;
    const float xs  = big ? -1.0f : x;
    const float rb  = __logf(-expm1f(xb));       // 1-exp(x) near 0: need expm1
    const float rs  = __logf(1.0f - __expf(xs)); // 1-exp(x) in [0.5,1): exact enough
    return big ? rb : rs;
}

// Full 36-term loss contribution of one row.
__device__ __forceinline__ float element_loss(
    float ab0, float ab1, float ba0, float ba1,
    float bc0, float bc1, float cb0, float cb1,
    float ac0, float ac1, float ca0, float ca1)
{
    // pair AB
    const float lab0 = log1mexp_f(ab0), lba0 = log1mexp_f(ba0);
    const float lab1 = log1mexp_f(ab1), lba1 = log1mexp_f(ba1);
    const float A00 = ab0 + lba0;
    const float A10 = lab0 + ba0;
    const float A20 = ab0 + ba0;
    const float A01 = ab1 + lba1;
    const float A11 = lab1 + ba1;
    const float A21 = ab1 + ba1;
    const float A31 = lab1 + lba1;
    // pair BC
    const float lbc0 = log1mexp_f(bc0), lcb0 = log1mexp_f(cb0);
    const float lbc1 = log1mexp_f(bc1), lcb1 = log1mexp_f(cb1);
    const float B00 = bc0 + lcb0;
    const float B10 = lbc0 + cb0;
    const float B20 = bc0 + cb0;
    const float B01 = bc1 + lcb1;
    const float B11 = lbc1 + cb1;
    const float B21 = bc1 + cb1;
    const float B31 = lbc1 + lcb1;
    // pair AC
    const float lac0 = log1mexp_f(ac0), lca0 = log1mexp_f(ca0);
    const float lac1 = log1mexp_f(ac1), lca1 = log1mexp_f(ca1);
    const float C01 = ac1 + lca1;
    const float C11 = lac1 + ca1;
    const float C21 = ac1 + ca1;
    const float C31 = lac1 + lca1;
    const float LC0 = log1mexp_f(ac0 + lca0);
    const float LC1 = log1mexp_f(lac0 + ca0);
    const float LC2 = log1mexp_f(ac0 + ca0);

    float acc = 0.0f;
#define RELU_ACC(t) acc += fmaxf(0.0f, (t))
    // LOSS_RECIPE (14)
    RELU_ACC(A00 + B01 - C01);  RELU_ACC(A00 + B21 - C01);
    RELU_ACC(A10 + B11 - C11);  RELU_ACC(A10 + B21 - C11);
    RELU_ACC(A20 + B01 - C01);  RELU_ACC(A20 + B11 - C11);
    RELU_ACC(A20 + B21 - C21);  RELU_ACC(A20 + B31 - C31);
    RELU_ACC(A01 + B00 - C01);  RELU_ACC(A01 + B20 - C01);
    RELU_ACC(A11 + B10 - C11);  RELU_ACC(A11 + B20 - C11);
    RELU_ACC(A21 + B20 - C21);  RELU_ACC(A31 + B20 - C31);
    // NEG_LOSS_RECIPE (22)
    RELU_ACC(A00 + B01 - LC1);  RELU_ACC(A00 + B01 - LC2);
    RELU_ACC(A00 + B21 - LC1);  RELU_ACC(A00 + B21 - LC2);
    RELU_ACC(A10 + B11 - LC0);  RELU_ACC(A10 + B11 - LC2);
    RELU_ACC(A10 + B21 - LC0);  RELU_ACC(A10 + B21 - LC2);
    RELU_ACC(A20 + B01 - LC1);  RELU_ACC(A20 + B01 - LC2);
    RELU_ACC(A20 + B11 - LC0);  RELU_ACC(A20 + B11 - LC2);
    RELU_ACC(A01 + B00 - LC1);  RELU_ACC(A01 + B00 - LC2);
    RELU_ACC(A01 + B20 - LC1);  RELU_ACC(A01 + B20 - LC2);
    RELU_ACC(A11 + B10 - LC0);  RELU_ACC(A11 + B10 - LC2);
    RELU_ACC(A11 + B20 - LC0);  RELU_ACC(A11 + B20 - LC2);
    RELU_ACC(A20 + B31 - LC2);  RELU_ACC(A31 + B20 - LC2);
#undef RELU_ACC
    return acc;
}

__device__ __forceinline__ float elem2_from(const float4& ab, const float4& ba,
                                            const float4& bc, const float4& cb,
                                            const float4& ac, const float4& ca)
{
    return element_loss(ab.x, ab.y, ba.x, ba.y, bc.x, bc.y, cb.x, cb.y,
                        ac.x, ac.y, ca.x, ca.y)
         + element_loss(ab.z, ab.w, ba.z, ba.w, bc.z, bc.w, cb.z, cb.w,
                        ac.z, ac.w, ca.z, ca.w);
}

// Issue one async b128 DMA: per lane, global memory at base plus byte offset voff into LDS at ldsOff.
// GVS addressing: SGPR64 base + VGPR 32-bit byte offset. Tracked by ASYNCcnt.
__device__ __forceinline__ void async_tile_b128(unsigned ldsOff, unsigned voff,
                                                const float* base)
{
    asm volatile("global_load_async_to_lds_b128 %0, %1, %2"
                 :: "v"(ldsOff), "v"(voff), "s"(base)
                 : "memory");
}

// Stage 1: async-LDS double-buffered pipeline over full tiles, grid-stride.
__global__ __launch_bounds__(WG_THREADS)
void BoxLoss_partial_kernel(const float* __restrict__ AB, const float* __restrict__ BA,
                            const float* __restrict__ BC, const float* __restrict__ CB,
                            const float* __restrict__ AC, const float* __restrict__ CA,
                            float* __restrict__ partial,
                            long long npairs, long long fullTiles, long long tailRow)
{
    // [buffer][array][thread] : 2 * 6 * 256 * 16 B = 48 KB of the 320 KB WGP LDS
    __shared__ float4 sBuf[2][6][TILE];

    const int tid = threadIdx.x;
    float acc = 0.0f;

#define ISSUE_TILE(bufi, t)                                                    \
    do {                                                                       \
        const unsigned voff = (unsigned)((((t) * (long long)TILE) + tid) * 16);\
        async_tile_b128((unsigned)(size_t)&sBuf[(bufi)][0][tid], voff, AB);    \
        async_tile_b128((unsigned)(size_t)&sBuf[(bufi)][1][tid], voff, BA);    \
        async_tile_b128((unsigned)(size_t)&sBuf[(bufi)][2][tid], voff, BC);    \
        async_tile_b128((unsigned)(size_t)&sBuf[(bufi)][3][tid], voff, CB);    \
        async_tile_b128((unsigned)(size_t)&sBuf[(bufi)][4][tid], voff, AC);    \
        async_tile_b128((unsigned)(size_t)&sBuf[(bufi)][5][tid], voff, CA);    \
    } while (0)

    long long t = (long long)blockIdx.x;
    const long long tstride = (long long)gridDim.x;
    int buf = 0;

    if (t < fullTiles) ISSUE_TILE(0, t);          // prologue: prime buffer 0

    for (; t < fullTiles; t += tstride) {
        const long long tn = t + tstride;
        if (tn < fullTiles) {
            ISSUE_TILE(buf ^ 1, tn);              // prefetch next tile via DMA
            // 12 outstanding max; <=6 left  => current tile's 6 are complete
            asm volatile("s_wait_asynccnt 0x6" ::: "memory");
        } else {
            asm volatile("s_wait_asynccnt 0x0" ::: "memory");
        }
        // Consume own lanes from LDS (ds_load_b128); no cross-wave sharing,
        // so no workgroup barrier is required.
        const float4 ab = sBuf[buf][0][tid];
        const float4 ba = sBuf[buf][1][tid];
        const float4 bc = sBuf[buf][2][tid];
        const float4 cb = sBuf[buf][3][tid];
        const float4 ac = sBuf[buf][4][tid];
        const float4 ca = sBuf[buf][5][tid];
        acc += elem2_from(ab, ba, bc, cb, ac, ca);
        buf ^= 1;
    }
#undef ISSUE_TILE

    // Remainder float4s (partial last tile) via direct loads, block 0 only.
    if (blockIdx.x == 0) {
        const long long g = fullTiles * TILE + tid;
        if (g < npairs) {
            const float4* AB4 = (const float4*)AB; const float4* BA4 = (const float4*)BA;
            const float4* BC4 = (const float4*)BC; const float4* CB4 = (const float4*)CB;
            const float4* AC4 = (const float4*)AC; const float4* CA4 = (const float4*)CA;
            acc += elem2_from(AB4[g], BA4[g], BC4[g], CB4[g], AC4[g], CA4[g]);
        }
        // Odd-N tail row handled by exactly one thread.
        if (tailRow >= 0 && tid == 0) {
            const long long r = 2 * tailRow;
            acc += element_loss(AB[r], AB[r + 1], BA[r], BA[r + 1],
                                BC[r], BC[r + 1], CB[r], CB[r + 1],
                                AC[r], AC[r + 1], CA[r], CA[r + 1]);
        }
    }

    // wave32 shuffle reduce, then cross-wave via LDS. Deterministic.
    float v = acc;
    #pragma unroll
    for (int off = 16; off > 0; off >>= 1)
        v += __shfl_down(v, off, 32);

    __shared__ float wsum[WG_THREADS / 32];
    const int lane = threadIdx.x & 31;
    const int wid  = threadIdx.x >> 5;
    if (lane == 0) wsum[wid] = v;
    __syncthreads();
    if (wid == 0) {
        v = (lane < (WG_THREADS / 32)) ? wsum[lane] : 0.0f;
        #pragma unroll
        for (int off = (WG_THREADS / 64); off > 0; off >>= 1)
            v += __shfl_down(v, off, 32);
        if (lane == 0) partial[blockIdx.x] = v;
    }
}

// Stage 2: single block, fixed-tree double-precision reduction (deterministic).
__global__ __launch_bounds__(256)
void BoxLoss_reduce_kernel(const float* __restrict__ partial, int n, float* __restrict__ out)
{
    __shared__ double s[256];
    double a = 0.0;
    for (int i = threadIdx.x; i < n; i += 256)
        a += (double)partial[i];
    s[threadIdx.x] = a;
    __syncthreads();
    #pragma unroll
    for (int w = 128; w > 0; w >>= 1) {
        if ((int)threadIdx.x < w) s[threadIdx.x] += s[threadIdx.x + w];
        __syncthreads();
    }
    if (threadIdx.x == 0) out[0] = (float)s[0];
}

extern "C" void kernel_launch(void* const* d_in, const int* in_sizes, int n_in,
                              void* d_out, int out_size, void* d_ws, size_t ws_size,
                              hipStream_t stream) {
    // Inputs: vol_AB, vol_BA, vol_BC, vol_CB, vol_AC, vol_CA, then three
    // rel_id arrays which the reference never uses (skipped entirely).
    const float* AB = (const float*)d_in[0];
    const float* BA = (const float*)d_in[1];
    const float* BC = (const float*)d_in[2];
    const float* CB = (const float*)d_in[3];
    const float* AC = (const float*)d_in[4];
    const float* CA = (const float*)d_in[5];

    const long long N         = (long long)in_sizes[0] / 2;  // rows of (N,2)
    const long long npairs    = N / 2;                       // float4 units
    const long long fullTiles = npairs / TILE;               // async-pipelined
    const long long tailRow   = (N & 1) ? (N - 1) : -1;

    float* partial = (float*)d_ws;                           // WG_BLOCKS floats

    BoxLoss_partial_kernel<<<WG_BLOCKS, WG_THREADS, 0, stream>>>(
        AB, BA, BC, CB, AC, CA, partial, npairs, fullTiles, tailRow);
    BoxLoss_reduce_kernel<<<1, 256, 0, stream>>>(partial, WG_BLOCKS, (float*)d_out);
}